// FilterDetections_27504970564178
// MI455X (gfx1250) — compile-verified
//
#include <hip/hip_runtime.h>
#include <hip/hip_bf16.h>
#include <stdint.h>

#define MAX_DET      300
#define NMS_THRESH   0.5f
#define SCORE_THRESH 0.05f
#define NMS_BLOCK    1024   // 32 wave32 waves
#define TILE         2048   // boxes per LDS tile (32 KB per buffer), 2 boxes/thread
#define NBUF         3      // triple-buffered TDM pipeline: 1 barrier per tile
#define TOPK_BLOCK   256

// ---- CDNA5 Tensor Data Mover availability (device pass only) ----
#if defined(__HIP_DEVICE_COMPILE__) && __has_builtin(__builtin_amdgcn_tensor_load_to_lds)
#define K_TDM 1
#else
#define K_TDM 0
#endif

typedef unsigned int v4u __attribute__((ext_vector_type(4)));
typedef int          v4i __attribute__((ext_vector_type(4)));
typedef int          v8i __attribute__((ext_vector_type(8)));

template <int Ncnt>
__device__ __forceinline__ void tdm_wait() {
#if defined(__HIP_DEVICE_COMPILE__) && __has_builtin(__builtin_amdgcn_s_wait_tensorcnt)
    __builtin_amdgcn_s_wait_tensorcnt(Ncnt);
#endif
}

// 1-D TDM load: tile_dw dwords from gptr into LDS byte offset lds_off.
// tensor_dw = dwords remaining in the source (OOB reads return zero).
__device__ __forceinline__ void tdm_load_1d(uint32_t lds_off, const void* gptr,
                                            uint32_t tile_dw, uint32_t tensor_dw) {
#if K_TDM
    uint64_t ga = (uint64_t)(uintptr_t)gptr;
    v4u g0;
    g0.x = 1u;                                                // count=1 (valid), user mode
    g0.y = lds_off;                                           // lds_addr
    g0.z = (uint32_t)ga;                                      // global_addr[31:0]
    g0.w = (uint32_t)((ga >> 32) & 0x01FFFFFFu) | (2u << 30); // addr[56:32] | type=2
    v8i g1;
    g1[0] = (int)(2u << 16);                                  // data_size=4B, wg_mask=0
    g1[1] = (int)((tensor_dw & 0xFFFFu) << 16);               // tensor_dim0[15:0]
    g1[2] = (int)(((tensor_dw >> 16) & 0xFFFFu) | (1u << 16));// dim0 hi16 | tensor_dim1=1
    g1[3] = (int)((tile_dw & 0xFFFFu) << 16);                 // tile_dim0
    g1[4] = 0;                                                // tile_dim1=0, tile_dim2=0
    g1[5] = (int)tensor_dw;                                   // tensor_dim0_stride lo32
    g1[6] = 0;
    g1[7] = 0;
    v4i z4 = {0, 0, 0, 0};
#if __clang_major__ >= 23
    v8i z8 = {0, 0, 0, 0, 0, 0, 0, 0};
    __builtin_amdgcn_tensor_load_to_lds(g0, g1, z4, z4, z8, 0);
#else
    __builtin_amdgcn_tensor_load_to_lds(g0, g1, z4, z4, 0);
#endif
#else
    (void)lds_off; (void)gptr; (void)tile_dw; (void)tensor_dw;
#endif
}

// wave32 argmax reduction; prefers lowest index on ties (matches jnp.argmax)
__device__ __forceinline__ void waveArgmax(float& v, int& i) {
#pragma unroll
    for (int off = 16; off > 0; off >>= 1) {
        float ov = __shfl_down(v, off, 32);
        int   oi = __shfl_down(i, off, 32);
        if (ov > v || (ov == v && oi < i)) { v = ov; i = oi; }
    }
}

// ---------------------------------------------------------------------------
// Kernel 1: one workgroup per (image, class); grid = (C, B).
// Scores resident in LDS (200 KB). Boxes streamed through triple-buffered LDS
// tiles via the Tensor Data Mover (one workgroup barrier per tile). The
// suppression sweep is fused with the argmax for the NEXT iteration.
// ---------------------------------------------------------------------------
__global__ __launch_bounds__(NMS_BLOCK)
void fd_nms_kernel(const float* __restrict__ boxes, const float* __restrict__ cls,
                   int N,
                   float* __restrict__ ws_score, int* __restrict__ ws_idx) {
    extern __shared__ char smem[];
    float* ss = (float*)smem;                                // N scores
    const uint32_t scoreBytes = (((uint32_t)N * 4u) + 15u) & ~15u;
    float4* bbuf = (float4*)(smem + scoreBytes);             // NBUF * TILE float4 tiles

    __shared__ float s_rv[32];
    __shared__ int   s_ri[32];
    __shared__ float s_box[4];
    __shared__ int   s_flag;

    const int   tid = (int)threadIdx.x;
    const int   c   = (int)blockIdx.x;
    const int   b   = (int)blockIdx.y;
    const int   C   = (int)gridDim.x;
    const int   bc  = b * C + c;
    const float NEG_INF = -__builtin_inff();

    // threshold-filtered scores into LDS
    const float* clsRow = cls + (size_t)b * N * C + c;
    for (int i = tid; i < N; i += NMS_BLOCK) {
        if (i + 4 * NMS_BLOCK < N)
            __builtin_prefetch(&clsRow[(size_t)(i + 4 * NMS_BLOCK) * C], 0, 1);
        float v = clsRow[(size_t)i * C];
        ss[i] = (v > SCORE_THRESH) ? v : NEG_INF;
    }
    __syncthreads();

    const float4*  gbox     = (const float4*)boxes + (size_t)b * N;
    const int      T        = (N + TILE - 1) / TILE;
    const uint32_t buf_off0 = (uint32_t)(uintptr_t)(void*)bbuf;   // LDS byte offset

    float* ows  = ws_score + (size_t)bc * MAX_DET;
    int*   oidx = ws_idx   + (size_t)bc * MAX_DET;

    // ---- seed argmax (vectorized float4 LDS scan) ----
    {
        float bv = NEG_INF; int bi = 0;
        const float4* ss4 = (const float4*)ss;
        const int N4 = N >> 2;
        for (int i4 = tid; i4 < N4; i4 += NMS_BLOCK) {
            const float4 v = ss4[i4];
            const int base = i4 << 2;
            if (v.x > bv) { bv = v.x; bi = base; }
            if (v.y > bv) { bv = v.y; bi = base + 1; }
            if (v.z > bv) { bv = v.z; bi = base + 2; }
            if (v.w > bv) { bv = v.w; bi = base + 3; }
        }
        for (int i = (N4 << 2) + tid; i < N; i += NMS_BLOCK) {
            const float v = ss[i];
            if (v > bv) { bv = v; bi = i; }
        }
        waveArgmax(bv, bi);
        if ((tid & 31) == 0) { s_rv[tid >> 5] = bv; s_ri[tid >> 5] = bi; }
        __syncthreads();
    }

    for (int it = 0; it < MAX_DET; ++it) {
        // ---- final 32-entry reduce + selection bookkeeping ----
        if (tid < 32) {
            float v = s_rv[tid]; int ix = s_ri[tid];
            waveArgmax(v, ix);
            if (tid == 0) {
                const int sel = (v > NEG_INF) ? 1 : 0;
                s_flag   = sel;
                ows[it]  = sel ? v : NEG_INF;
                oidx[it] = ix;
                if (sel) {
                    const float4 bb = gbox[ix];
                    s_box[0] = bb.x; s_box[1] = bb.y; s_box[2] = bb.z; s_box[3] = bb.w;
                    ss[ix] = NEG_INF;
                }
            }
        }
        __syncthreads();

        if (!s_flag) {   // no finite candidate left: pad and stop (uniform)
            for (int k = it + 1 + tid; k < MAX_DET; k += NMS_BLOCK) {
                ows[k]  = NEG_INF;
                oidx[k] = 0;
            }
            break;
        }
        if (it + 1 == MAX_DET) break;   // last selection done; no sweep needed

        // ---- fused IoU-suppression + next-argmax sweep (TDM triple buffer) ----
        const float ax1 = s_box[0], ay1 = s_box[1], ax2 = s_box[2], ay2 = s_box[3];
        const float areaA = (ax2 - ax1) * (ay2 - ay1);

        if (K_TDM && tid < 32)  // prologue: tile 0 -> buffer 0
            tdm_load_1d(buf_off0, (const void*)gbox, TILE * 4u, (uint32_t)N * 4u);

        float bv = NEG_INF; int bi = 0;
        int cur = 0;
        for (int t = 0; t < T; ++t) {
            const int nxt = (cur == NBUF - 1) ? 0 : cur + 1;
            if (K_TDM && tid < 32) {
                if (t + 1 < T) {
                    const uint32_t off = buf_off0 + (uint32_t)nxt * (TILE * 16u);
                    const uint32_t rem = (uint32_t)(N - (t + 1) * TILE) * 4u;
                    tdm_load_1d(off, (const void*)(gbox + (size_t)(t + 1) * TILE),
                                TILE * 4u, rem);
                    tdm_wait<1>();   // oldest (tile t) complete
                } else {
                    tdm_wait<0>();
                }
            }
            __syncthreads();   // tile t visible; buf[nxt of next iter] fenced by this

#pragma unroll
            for (int k = 0; k < TILE / NMS_BLOCK; ++k) {
                const int j = t * TILE + k * NMS_BLOCK + tid;
                if (j < N) {
                    float4 bb;
                    if (K_TDM) bb = bbuf[(size_t)(cur * TILE + k * NMS_BLOCK + tid)];
                    else       bb = gbox[j];
                    const float x1 = fmaxf(ax1, bb.x), y1 = fmaxf(ay1, bb.y);
                    const float x2 = fminf(ax2, bb.z), y2 = fminf(ay2, bb.w);
                    const float inter = fmaxf(x2 - x1, 0.0f) * fmaxf(y2 - y1, 0.0f);
                    const float areaB = (bb.z - bb.x) * (bb.w - bb.y);
                    const float uni   = fmaxf(areaA + areaB - inter, 1e-8f);
                    float v = ss[j];
                    if (inter / uni > NMS_THRESH) { v = NEG_INF; ss[j] = NEG_INF; }
                    if (v > bv) { bv = v; bi = j; }   // j increases with t,k: lowest idx wins
                }
            }
            cur = nxt;
        }
        waveArgmax(bv, bi);
        if ((tid & 31) == 0) { s_rv[tid >> 5] = bv; s_ri[tid >> 5] = bi; }
        __syncthreads();   // also fences last tile's buffer before next prologue
    }
}

// ---------------------------------------------------------------------------
// Kernel 2: one workgroup per image. Global top-MAX_DET over C*MAX_DET
// per-class survivors, then gather boxes/other with -1 padding.
// ---------------------------------------------------------------------------
__global__ __launch_bounds__(TOPK_BLOCK)
void fd_topk_kernel(const float* __restrict__ boxes, const float* __restrict__ other,
                    const float* __restrict__ ws_score, const int* __restrict__ ws_idx,
                    int N, int C, int D, float* __restrict__ out, int B) {
    __shared__ float cs[16 * MAX_DET];          // up to C=16 classes
    __shared__ float s_rv[TOPK_BLOCK / 32];
    __shared__ int   s_ri[TOPK_BLOCK / 32];

    const int   b   = (int)blockIdx.x;
    const int   tid = (int)threadIdx.x;
    const float NEG_INF = -__builtin_inff();
    int M = C * MAX_DET;
    if (M > 16 * MAX_DET) M = 16 * MAX_DET;

    const float* wsS = ws_score + (size_t)b * C * MAX_DET;
    const int*   wsI = ws_idx   + (size_t)b * C * MAX_DET;
    for (int i = tid; i < M; i += TOPK_BLOCK) cs[i] = wsS[i];
    __syncthreads();

    float* ob = out;
    float* os = out + (size_t)B * MAX_DET * 4;
    int*   ol = (int*)(os + (size_t)B * MAX_DET);
    float* oo = (float*)(ol + (size_t)B * MAX_DET);

    for (int m = 0; m < MAX_DET; ++m) {
        float bv = NEG_INF; int bi = 0;
        for (int i = tid; i < M; i += TOPK_BLOCK) {
            const float v = cs[i];
            if (v > bv) { bv = v; bi = i; }
        }
        waveArgmax(bv, bi);
        if ((tid & 31) == 0) { s_rv[tid >> 5] = bv; s_ri[tid >> 5] = bi; }
        __syncthreads();
        if (tid == 0) {
            float v = s_rv[0]; int ix = s_ri[0];
#pragma unroll
            for (int w = 1; w < TOPK_BLOCK / 32; ++w)
                if (s_rv[w] > v || (s_rv[w] == v && s_ri[w] < ix)) { v = s_rv[w]; ix = s_ri[w]; }
            const size_t row = (size_t)b * MAX_DET + m;
            if (v > NEG_INF) {
                const int srcIdx = wsI[ix];
                const int lab    = ix / MAX_DET;
                const float* bp = boxes + ((size_t)b * N + srcIdx) * 4;
                ob[row * 4 + 0] = bp[0]; ob[row * 4 + 1] = bp[1];
                ob[row * 4 + 2] = bp[2]; ob[row * 4 + 3] = bp[3];
                os[row] = v;
                ol[row] = lab;
                const float* op = other + ((size_t)b * N + srcIdx) * D;
                for (int d = 0; d < D; ++d) oo[row * D + d] = op[d];
                cs[ix] = NEG_INF;
            } else {
                ob[row * 4 + 0] = -1.0f; ob[row * 4 + 1] = -1.0f;
                ob[row * 4 + 2] = -1.0f; ob[row * 4 + 3] = -1.0f;
                os[row] = -1.0f;
                ol[row] = -1;
                for (int d = 0; d < D; ++d) oo[row * D + d] = -1.0f;
            }
        }
        __syncthreads();
    }
}

// ---------------------------------------------------------------------------
extern "C" void kernel_launch(void* const* d_in, const int* in_sizes, int n_in,
                              void* d_out, int out_size, void* d_ws, size_t ws_size,
                              hipStream_t stream) {
    const float* boxes = (const float*)d_in[0];
    const float* cls   = (const float*)d_in[1];
    const float* other = (const float*)d_in[2];

    // Infer shapes: out = B*300*(4+1+1+D) with D=2 -> 2400/img.
    const int B = out_size / (MAX_DET * 8);
    const int N = in_sizes[0] / (4 * B);
    const int C = in_sizes[1] / (B * N);
    const int D = in_sizes[2] / (B * N);

    float* ws_score = (float*)d_ws;
    int*   ws_idx   = (int*)(ws_score + (size_t)B * C * MAX_DET);

    const size_t scoreBytes = (((size_t)N * 4) + 15) & ~(size_t)15;
    const size_t dynBytes   = scoreBytes + (size_t)NBUF * TILE * 16;  // ~296 KB < 320 KB WGP LDS
    hipFuncSetAttribute((const void*)fd_nms_kernel,
                        hipFuncAttributeMaxDynamicSharedMemorySize, (int)dynBytes);

    dim3 nmsGrid((unsigned)C, (unsigned)B, 1);
    fd_nms_kernel<<<nmsGrid, NMS_BLOCK, dynBytes, stream>>>(boxes, cls, N,
                                                            ws_score, ws_idx);
    fd_topk_kernel<<<B, TOPK_BLOCK, 0, stream>>>(boxes, other, ws_score, ws_idx,
                                                 N, C, D, (float*)d_out, B);
}